// generatePairwiseF_keras1_5927054868544
// MI455X (gfx1250) — compile-verified
//
#include <hip/hip_runtime.h>
#include <stdint.h>

// ---------------------------------------------------------------------------
// Pairwise concat:  out[i][j][:] = [ x[i] | (x[i]+x[j])/2 | x[j] ]
//   x:   (768, 64)  fp32   = 196 KB   (fits entirely in LDS: 320 KB/WGP)
//   out: (768, 768, 192) fp32 = 432 MB  -> pure store-bandwidth problem
// Roofline @ 23.3 TB/s: ~19 us. Strategy:
//   * One TDM tensor_load_to_lds stages the whole input per workgroup
//     (single descriptor, tile == whole 64x768 fp32 tensor).
//   * Inner loop is the minimal stream: ds_load_b128 + 4x v_fmac_f32 +
//     non-temporal global_store_b128 (432MB > 192MB L2, write-once).
//   * Per-thread section select folded into loop-invariant FMA coeffs:
//     v = acc0 + b*xj,  (acc0,b) in {(xi,0), (xi/2,1/2), (0,1)}.
// ---------------------------------------------------------------------------

typedef float        f32x4 __attribute__((ext_vector_type(4)));
typedef unsigned int u32x4 __attribute__((ext_vector_type(4)));
typedef int          i32x4 __attribute__((ext_vector_type(4)));
typedef int          i32x8 __attribute__((ext_vector_type(8)));

#define L_DIM 768
#define N_DIM 64
#define OUTF  192            // 3 * N_DIM floats per (i,j) row
#define CH    48             // float4 chunks per output row (192/4)
#define JT    16             // j's handled per pass (768 threads / 48)
#define BLOCK 768

__global__ __launch_bounds__(BLOCK) void pairwise_concat_kernel(
    const float* __restrict__ x, float* __restrict__ out) {
  extern __shared__ float smem[];  // full x: 768*64 floats = 196608 B

  // ---- Stage the entire input tensor into LDS with one TDM descriptor ----
  if (threadIdx.x < 32) {   // one wave issues the DMA (EXEC ignored by TDM)
    const uint64_t ga       = (uint64_t)(const void*)x;
    const uint32_t lds_base = (uint32_t)__builtin_amdgcn_groupstaticsize();

    u32x4 g0;
    g0.x = 1u;                                                // count=1
    g0.y = lds_base;                                          // lds_addr
    g0.z = (uint32_t)(ga & 0xFFFFFFFFu);                      // global_addr lo
    g0.w = (uint32_t)((ga >> 32) & 0x01FFFFFFu) | (2u << 30); // addr hi | type=2

    const uint64_t d0s = (uint64_t)N_DIM;          // dim0 stride (elements)
    const uint64_t d1s = (uint64_t)N_DIM * L_DIM;  // dim1 stride (elements)
    i32x8 g1;
    g1[0] = (int)(2u << 16);                                   // data_size=4B
    g1[1] = (int)((uint32_t)(N_DIM & 0xFFFF) << 16);           // tensor_dim0 lo16
    g1[2] = (int)(((uint32_t)N_DIM >> 16) |
                  ((uint32_t)(L_DIM & 0xFFFF) << 16));         // dim0 hi | dim1 lo
    g1[3] = (int)(((uint32_t)L_DIM >> 16) |
                  ((uint32_t)N_DIM << 16));                    // dim1 hi | tile_dim0
    g1[4] = (int)(uint32_t)L_DIM;                              // tile_dim1 (tile_dim2=0)
    g1[5] = (int)(uint32_t)(d0s & 0xFFFFFFFFu);                // dim0_stride lo32
    g1[6] = (int)((uint32_t)((d0s >> 32) & 0xFFFFu) |
                  (uint32_t)((d1s & 0xFFFFu) << 16));          // d0s hi16 | d1s lo16
    g1[7] = (int)(uint32_t)((d1s >> 16) & 0xFFFFFFFFu);        // d1s hi32

    const i32x4 gz4 = {0, 0, 0, 0};            // groups 2/3 unused (2D tensor)
    const i32x8 gz8 = {0, 0, 0, 0, 0, 0, 0, 0};
    __builtin_amdgcn_tensor_load_to_lds(g0, g1, gz4, gz4, gz8, 0);
    __builtin_amdgcn_s_wait_tensorcnt(0);
  }
  __syncthreads();

  // ---- Stream the 432 MB of output -------------------------------------
  const int i   = blockIdx.x;          // one i per block
  const int t   = threadIdx.x;
  const int c   = t % CH;              // float4 chunk within the 192-f row
  const int jl  = t / CH;              // 0..15: j within a pass
  const int sec = c >> 4;              // 0: x_i, 1: avg, 2: x_j
  const int f   = (c & 15) * 4;        // feature offset inside section

  const f32x4 xi4 = *(const f32x4*)&smem[i * N_DIM + f];

  // Loop-invariant coefficients: v = acc0 + bco * xj4
  const float bco  = (sec == 0) ? 0.0f : ((sec == 1) ? 0.5f : 1.0f);
  const f32x4 zero = {0.0f, 0.0f, 0.0f, 0.0f};
  const f32x4 acc0 = (sec == 0) ? xi4 : ((sec == 1) ? xi4 * 0.5f : zero);

  float* __restrict__ obase = out + (size_t)i * L_DIM * OUTF + (size_t)c * 4;
  const float* __restrict__ lbase = &smem[jl * N_DIM + f];

#pragma unroll 8
  for (int p = 0; p < L_DIM / JT; ++p) {
    const f32x4 xj4 = *(const f32x4*)(lbase + p * (JT * N_DIM)); // ds_load_b128
    const f32x4 v   = acc0 + bco * xj4;                          // 4x v_fmac_f32
    __builtin_nontemporal_store(                                 // b128 NT store
        v, (f32x4*)(obase + (size_t)(p * JT + jl) * OUTF));
  }
}

extern "C" void kernel_launch(void* const* d_in, const int* in_sizes, int n_in,
                              void* d_out, int out_size, void* d_ws, size_t ws_size,
                              hipStream_t stream) {
  (void)in_sizes; (void)n_in; (void)out_size; (void)d_ws; (void)ws_size;
  const float* x = (const float*)d_in[0];
  float* out     = (float*)d_out;

  const size_t shmem = (size_t)L_DIM * N_DIM * sizeof(float);  // 196608 B
  hipLaunchKernelGGL(pairwise_concat_kernel,
                     dim3(L_DIM), dim3(BLOCK), shmem, stream, x, out);
}